// GATLayer_25701084299552
// MI455X (gfx1250) — compile-verified
//
#include <hip/hip_runtime.h>
#include <math.h>

// GAT layer fused attention for MI455X (gfx1250), wave32 + WMMA f32.
//   x:   [B=4, C=4, N=2048, F=64] f32
//   adj: [N, N] i32
//   a1,a2: [F,1] f32
//   out = softmax(mask(leakyrelu(f1_i + f2_j))) @ x   per (b,c)

typedef __attribute__((ext_vector_type(2))) float v2f;
typedef __attribute__((ext_vector_type(8))) float v8f;

#define NROWS 2048
#define FDIM 64
#define NBC 16              // B*C
#define NEG_BIG (-9.0e15f)
#define ALPHA 0.1f

__device__ __forceinline__ float wave_reduce_add(float v) {
    #pragma unroll
    for (int off = 16; off > 0; off >>= 1) v += __shfl_xor(v, off, 32);
    return v;
}

__device__ __forceinline__ float lrelu_mask(float e, int a) {
    e = (e > 0.0f) ? e : ALPHA * e;
    return (a > 0) ? e : NEG_BIG;
}

// ---------------------------------------------------------------------------
// Kernel 1: f1[row] = x[row,:]·a1, f2[row] = x[row,:]·a2   (row = bc*2048+i)
// One wave per row, 2 features per lane.
// ---------------------------------------------------------------------------
__global__ void gat_proj_kernel(const float* __restrict__ x,
                                const float* __restrict__ a1,
                                const float* __restrict__ a2,
                                float* __restrict__ f1,
                                float* __restrict__ f2) {
    int wave = (int)((blockIdx.x * blockDim.x + threadIdx.x) >> 5);
    int lane = (int)(threadIdx.x & 31);
    if (wave >= NBC * NROWS) return;
    const float* xr = x + (size_t)wave * FDIM;
    float x0 = xr[lane], x1 = xr[lane + 32];
    float s1 = x0 * a1[lane] + x1 * a1[lane + 32];
    float s2 = x0 * a2[lane] + x1 * a2[lane + 32];
    s1 = wave_reduce_add(s1);
    s2 = wave_reduce_add(s2);
    if (lane == 0) { f1[wave] = s1; f2[wave] = s2; }
}

// ---------------------------------------------------------------------------
// Kernel 2: per-row softmax stats (max and 1/sum).
// One wave per (bc,row). Each lane consumes 4 columns per step via b128
// loads (int4 adj, float4 f2), does max-of-4 then a single rescale:
// 1.25 exps/element instead of 2. Final 5-step cross-lane (max,sum) merge.
// ---------------------------------------------------------------------------
__global__ void gat_stats_kernel(const int*   __restrict__ adj,
                                 const float* __restrict__ f1,
                                 const float* __restrict__ f2,
                                 float* __restrict__ rowmax,
                                 float* __restrict__ rowinv) {
    int wave = (int)((blockIdx.x * blockDim.x + threadIdx.x) >> 5);
    int lane = (int)(threadIdx.x & 31);
    if (wave >= NBC * NROWS) return;
    int bc = wave >> 11;          // / 2048
    int i  = wave & (NROWS - 1);
    float f1r = f1[wave];
    const float4* f2b4  = (const float4*)(f2 + bc * NROWS);
    const int4*   adjr4 = (const int4*)(adj + (size_t)i * NROWS);

    float m = -INFINITY, s = 0.0f;
    #pragma unroll 4
    for (int q = lane; q < NROWS / 4; q += 32) {
        int4   a4 = adjr4[q];
        float4 v4 = f2b4[q];
        float e0 = lrelu_mask(f1r + v4.x, a4.x);
        float e1 = lrelu_mask(f1r + v4.y, a4.y);
        float e2 = lrelu_mask(f1r + v4.z, a4.z);
        float e3 = lrelu_mask(f1r + v4.w, a4.w);
        float m4 = fmaxf(fmaxf(e0, e1), fmaxf(e2, e3));
        float nm = fmaxf(m, m4);              // finite after first step
        s = s * __expf(m - nm)
          + __expf(e0 - nm) + __expf(e1 - nm)
          + __expf(e2 - nm) + __expf(e3 - nm);
        m = nm;
    }
    #pragma unroll
    for (int off = 16; off > 0; off >>= 1) {
        float mo = __shfl_xor(m, off, 32);
        float so = __shfl_xor(s, off, 32);
        float nm = fmaxf(m, mo);
        s = s * __expf(m - nm) + so * __expf(mo - nm);
        m = nm;
    }
    if (lane == 0) { rowmax[wave] = m; rowinv[wave] = 1.0f / s; }
}

// ---------------------------------------------------------------------------
// Kernel 3: fused P-recompute + (P @ x) via V_WMMA_F32_16X16X4_F32.
// One wave per (bc, 32-row tile): TWO 16-row M tiles -> the B fragments
// (x tile, independent of M) are loaded once and feed 8 WMMAs per K-step
// (1 vmem B-load per WMMA). f2 scalars are wave-uniform -> SMEM loads;
// adj fetched as one b128 per lane per M tile.
//
// ISA layouts (CDNA5 ISA §7.12.2, f32 16x16x4):
//   A (16x4):  lane r=lane&15 is row M; VGPR0 holds K = 2*(lane>>4),
//              VGPR1 holds K = 2*(lane>>4)+1.
//   B (4x16):  lane n=lane&15 is col N; VGPR0 holds K = 2*(lane>>4),
//              VGPR1 holds K = 2*(lane>>4)+1.
//   C/D (16x16): VGPR v: lanes 0-15 -> M=v, lanes 16-31 -> M=v+8; N=lane&15.
// EXEC stays all-ones (no divergence, exact grid).
// ---------------------------------------------------------------------------
__global__ __launch_bounds__(32)
void gat_attn_mm_kernel(const float* __restrict__ x,
                        const int*   __restrict__ adj,
                        const float* __restrict__ f1,
                        const float* __restrict__ f2,
                        const float* __restrict__ rowmax,
                        const float* __restrict__ rowinv,
                        float* __restrict__ out) {
    int tile = (int)blockIdx.x;       // 0 .. NBC*(NROWS/32)-1
    int bc   = tile >> 6;             // / 64
    int i0   = (tile & 63) << 5;      // 32-row tile base
    int lane = (int)threadIdx.x;
    int r    = lane & 15;             // M row (A) / N col (B,C)
    int c    = lane >> 4;             // K half selector

    int rowg0 = bc * NROWS + i0 + r;        // M tile 0 row
    int rowg1 = rowg0 + 16;                 // M tile 1 row
    float vf1_0  = f1[rowg0],     vf1_1  = f1[rowg1];
    float vmax0  = rowmax[rowg0], vmax1  = rowmax[rowg1];
    float vinv0  = rowinv[rowg0], vinv1  = rowinv[rowg1];

    const float* xb    = x  + (size_t)bc * NROWS * FDIM;
    const int*   adjr0 = adj + (size_t)(i0 + r) * NROWS;
    const int*   adjr1 = adj + (size_t)(i0 + 16 + r) * NROWS;
    const float* f2b   = f2 + bc * NROWS;

    v8f acc00 = {}, acc01 = {}, acc02 = {}, acc03 = {};
    v8f acc10 = {}, acc11 = {}, acc12 = {}, acc13 = {};

    #pragma unroll 2
    for (int jj = 0; jj < NROWS; jj += 4) {
        // wave-uniform f2 scalars (SMEM) + per-half select
        float sf0 = f2b[jj + 0], sf1 = f2b[jj + 1];
        float sf2 = f2b[jj + 2], sf3 = f2b[jj + 3];
        float f2c0 = c ? sf2 : sf0;
        float f2c1 = c ? sf3 : sf1;

        // adjacency: one b128 per lane per M tile, select this half's pair
        int4 aj0 = *(const int4*)(adjr0 + jj);
        int4 aj1 = *(const int4*)(adjr1 + jj);
        int a00 = c ? aj0.z : aj0.x, a01 = c ? aj0.w : aj0.y;
        int a10 = c ? aj1.z : aj1.x, a11 = c ? aj1.w : aj1.y;

        // prefetch adj stream ~16 steps ahead (global_prefetch_b8)
        __builtin_prefetch(adjr0 + jj + 64, 0, 1);
        __builtin_prefetch(adjr1 + jj + 64, 0, 1);

        // A fragments: P[r][col0], P[r][col1] for both M tiles (exact f32)
        float p00 = __expf(lrelu_mask(vf1_0 + f2c0, a00) - vmax0) * vinv0;
        float p01 = __expf(lrelu_mask(vf1_0 + f2c1, a01) - vmax0) * vinv0;
        float p10 = __expf(lrelu_mask(vf1_1 + f2c0, a10) - vmax1) * vinv1;
        float p11 = __expf(lrelu_mask(vf1_1 + f2c1, a11) - vmax1) * vinv1;
        v2f A0 = {p00, p01};
        v2f A1 = {p10, p11};

        // B fragments: rows col0/col1 of x, features 16n + r; shared by
        // both M tiles (this is the amortized load)
        int col0 = jj + 2 * c;
        const float* xr0 = xb + (size_t)col0 * FDIM + r;
        const float* xr1 = xr0 + FDIM;
        v2f b0 = {xr0[0],  xr1[0]};
        v2f b1 = {xr0[16], xr1[16]};
        v2f b2 = {xr0[32], xr1[32]};
        v2f b3 = {xr0[48], xr1[48]};

        acc00 = __builtin_amdgcn_wmma_f32_16x16x4_f32(false, A0, false, b0,
                                                      (short)0, acc00, false, false);
        acc01 = __builtin_amdgcn_wmma_f32_16x16x4_f32(false, A0, false, b1,
                                                      (short)0, acc01, false, false);
        acc02 = __builtin_amdgcn_wmma_f32_16x16x4_f32(false, A0, false, b2,
                                                      (short)0, acc02, false, false);
        acc03 = __builtin_amdgcn_wmma_f32_16x16x4_f32(false, A0, false, b3,
                                                      (short)0, acc03, false, false);
        acc10 = __builtin_amdgcn_wmma_f32_16x16x4_f32(false, A1, false, b0,
                                                      (short)0, acc10, false, false);
        acc11 = __builtin_amdgcn_wmma_f32_16x16x4_f32(false, A1, false, b1,
                                                      (short)0, acc11, false, false);
        acc12 = __builtin_amdgcn_wmma_f32_16x16x4_f32(false, A1, false, b2,
                                                      (short)0, acc12, false, false);
        acc13 = __builtin_amdgcn_wmma_f32_16x16x4_f32(false, A1, false, b3,
                                                      (short)0, acc13, false, false);
    }

    // Write D tiles: VGPR v -> row (i0 + 16*t + v + 8*c), col = 16n + r
    float* ob = out + ((size_t)bc * NROWS + i0) * FDIM;
    #pragma unroll
    for (int v = 0; v < 8; v++) {
        int m0 = v + 8 * c;
        float* orow0 = ob + (size_t)m0 * FDIM + r;
        orow0[0]  = acc00[v];
        orow0[16] = acc01[v];
        orow0[32] = acc02[v];
        orow0[48] = acc03[v];
        float* orow1 = orow0 + 16 * FDIM;
        orow1[0]  = acc10[v];
        orow1[16] = acc11[v];
        orow1[32] = acc12[v];
        orow1[48] = acc13[v];
    }
}

// ---------------------------------------------------------------------------
extern "C" void kernel_launch(void* const* d_in, const int* in_sizes, int n_in,
                              void* d_out, int out_size, void* d_ws, size_t ws_size,
                              hipStream_t stream) {
    const float* x   = (const float*)d_in[0];   // [4,4,2048,64]
    const int*   adj = (const int*)d_in[1];     // [2048,2048]
    const float* a1  = (const float*)d_in[2];   // [64,1]
    const float* a2  = (const float*)d_in[3];   // [64,1]
    float* out = (float*)d_out;

    const int totalRows = NBC * NROWS;          // 32768
    float* ws   = (float*)d_ws;                 // 4 * 32768 * 4B = 512 KB
    float* f1   = ws;
    float* f2   = ws + totalRows;
    float* rmax = ws + 2 * totalRows;
    float* rinv = ws + 3 * totalRows;

    // 8 waves per 256-thread block, one wave per row
    dim3 blk(256), grd(totalRows / 8);
    gat_proj_kernel<<<grd, blk, 0, stream>>>(x, a1, a2, f1, f2);
    gat_stats_kernel<<<grd, blk, 0, stream>>>(adj, f1, f2, rmax, rinv);

    // one wave per (bc, 32-row tile): 16 * 64 = 1024 blocks of 32 threads
    dim3 blk2(32), grd2(NBC * (NROWS / 32));
    gat_attn_mm_kernel<<<grd2, blk2, 0, stream>>>(x, adj, f1, f2, rmax, rinv, out);
}